// TemporalFusion_4002909520203
// MI455X (gfx1250) — compile-verified
//
#include <hip/hip_runtime.h>

// BEV temporal alignment (bilinear ego-motion warp) for MI455X / gfx1250.
// Bandwidth-bound: ~0.3 GFLOP vs ~0.4 GB HBM traffic (floor ~18us @ 23.3TB/s).
// Strategy: per-batch trig hoisted to a 4-thread setup kernel (kills the
// Payne-Hanek VALU bloat in the hot loop); one thread per (b,h,w,c4) handles
// all 5 time slices so the 4 corner addresses are computed once and the per-t
// delta folds into the b128 load's 24-bit immediate offset; NT stores keep
// the 192MB L2 free for the gather-reusable source. WMMA inapplicable: the
// blend's "B operand" (corner vectors) varies per output pixel (= per row).

typedef float v4f __attribute__((ext_vector_type(4)));

namespace {
constexpr int   B_ = 4, H_ = 200, W_ = 200, T_ = 5, C_ = 64;
constexpr int   C4 = C_ / 4;                          // 16 float4 per slice
constexpr float MAP_RANGE = 200.0f;
constexpr unsigned N_THREADS = (unsigned)B_ * H_ * W_ * C4;   // 2,560,000
}

// One thread per batch: c, s, tx, ty -> ws[b] (float4)
__global__ void bev_setup_kernel(const float* __restrict__ ego,
                                 float* __restrict__ params)
{
    int b = threadIdx.x;
    if (b >= B_) return;
    float dxn  = ego[b * 3 + 0] / (MAP_RANGE * 0.5f);
    float dyn  = ego[b * 3 + 1] / (MAP_RANGE * 0.5f);
    float dyaw = ego[b * 3 + 2];
    float c = cosf(dyaw);
    float s = sinf(dyaw);
    float tx = -(c * dxn + s * dyn);
    float ty =  s * dxn - c * dyn;
    v4f p = {c, s, tx, ty};
    ((v4f*)params)[b] = p;
}

__global__ __launch_bounds__(256)
void bev_warp_kernel(const float* __restrict__ raw,
                     const float* __restrict__ params,
                     float* __restrict__ out)
{
    unsigned tid = blockIdx.x * 256u + threadIdx.x;
    if (tid >= N_THREADS) return;

    // 16 consecutive lanes cover one channel vector -> coalesced b128
    unsigned c4  = tid & 15u;
    unsigned pix = tid >> 4;                 // b*H*W + h*W + w
    unsigned w   = pix % (unsigned)W_;
    unsigned hw  = pix / (unsigned)W_;
    unsigned h   = hw % (unsigned)H_;
    unsigned b   = hw / (unsigned)H_;

    v4f p = ((const v4f*)params)[b];
    float c = p.x, s = p.y, tx = p.z, ty = p.w;

    float xs = (2.0f * (float)w + 1.0f) / (float)W_ - 1.0f;
    float ys = (2.0f * (float)h + 1.0f) / (float)H_ - 1.0f;
    float gx =  c * xs + s * ys + tx;
    float gy = -s * xs + c * ys + ty;
    float ix = ((gx + 1.0f) * (float)W_ - 1.0f) * 0.5f;
    float iy = ((gy + 1.0f) * (float)H_ - 1.0f) * 0.5f;

    float ix0f = floorf(ix);
    float iy0f = floorf(iy);
    float wx1 = ix - ix0f, wx0 = 1.0f - wx1;
    float wy1 = iy - iy0f, wy0 = 1.0f - wy1;
    float ix1f = ix0f + 1.0f;
    float iy1f = iy0f + 1.0f;

    const float Wm1 = (float)(W_ - 1), Hm1 = (float)(H_ - 1);
    float vx0 = (ix0f >= 0.0f && ix0f <= Wm1) ? 1.0f : 0.0f;
    float vx1 = (ix1f >= 0.0f && ix1f <= Wm1) ? 1.0f : 0.0f;
    float vy0 = (iy0f >= 0.0f && iy0f <= Hm1) ? 1.0f : 0.0f;
    float vy1 = (iy1f >= 0.0f && iy1f <= Hm1) ? 1.0f : 0.0f;

    int x0 = (int)fminf(fmaxf(ix0f, 0.0f), Wm1);
    int x1 = (int)fminf(fmaxf(ix1f, 0.0f), Wm1);
    int y0 = (int)fminf(fmaxf(iy0f, 0.0f), Hm1);
    int y1 = (int)fminf(fmaxf(iy1f, 0.0f), Hm1);

    float w00 = (wy0 * wx0) * (vy0 * vx0);
    float w01 = (wy0 * wx1) * (vy0 * vx1);
    float w10 = (wy1 * wx0) * (vy1 * vx0);
    float w11 = (wy1 * wx1) * (vy1 * vx1);

    const v4f* __restrict__ in4 = (const v4f*)raw;
    v4f* __restrict__ out4 = (v4f*)out;

    // corner base indices (float4 units); per-t delta (t*C4*16B) folds into
    // the global_load_b128 immediate offset in the unrolled loop below
    unsigned base_b = b * (unsigned)(H_ * W_ * T_ * C4);
    unsigned i00 = base_b + ((unsigned)y0 * W_ + (unsigned)x0) * (T_ * C4) + c4;
    unsigned i01 = base_b + ((unsigned)y0 * W_ + (unsigned)x1) * (T_ * C4) + c4;
    unsigned i10 = base_b + ((unsigned)y1 * W_ + (unsigned)x0) * (T_ * C4) + c4;
    unsigned i11 = base_b + ((unsigned)y1 * W_ + (unsigned)x1) * (T_ * C4) + c4;
    unsigned ob  = pix * (unsigned)(T_ * C4) + c4;

#pragma unroll
    for (int t = 0; t < T_ - 1; ++t) {
        v4f g00 = in4[i00 + t * C4];
        v4f g01 = in4[i01 + t * C4];
        v4f g10 = in4[i10 + t * C4];
        v4f g11 = in4[i11 + t * C4];
        // left-to-right sum order matches the reference expression
        v4f acc = w00 * g00;
        acc = acc + w01 * g01;
        acc = acc + w10 * g10;
        acc = acc + w11 * g11;
        __builtin_nontemporal_store(acc, out4 + ob + t * C4);
    }

    // t = T-1: pass-through (same linear index in and out), read-once ->
    // nontemporal both directions
    v4f cur = __builtin_nontemporal_load(in4 + ob + (T_ - 1) * C4);
    __builtin_nontemporal_store(cur, out4 + ob + (T_ - 1) * C4);
}

extern "C" void kernel_launch(void* const* d_in, const int* in_sizes, int n_in,
                              void* d_out, int out_size, void* d_ws, size_t ws_size,
                              hipStream_t stream) {
    (void)in_sizes; (void)n_in; (void)out_size; (void)ws_size;
    const float* raw = (const float*)d_in[0];   // (4,200,200,5,64) f32
    const float* ego = (const float*)d_in[1];   // (4,3) f32
    float* out    = (float*)d_out;              // (4,200,200,5,64) f32
    float* params = (float*)d_ws;               // 4 x float4 scratch

    bev_setup_kernel<<<1, 32, 0, stream>>>(ego, params);

    dim3 blk(256);
    dim3 grid((N_THREADS + 255u) / 256u);       // 10,000 blocks x 8 wave32
    bev_warp_kernel<<<grid, blk, 0, stream>>>(raw, params, out);
}